// BipartiteSAGEEncoder_64854006170163
// MI455X (gfx1250) — compile-verified
//
#include <hip/hip_runtime.h>

typedef float v2f __attribute__((ext_vector_type(2)));
typedef float v8f __attribute__((ext_vector_type(8)));

#define WPAD 132   // padded LDS row stride (floats): conflict-free b64 reads across wave32

// ---------------- zero fill ----------------
__global__ void zero_f32(float* __restrict__ p, long n) {
    long i = (long)blockIdx.x * blockDim.x + threadIdx.x;
    long n4 = n >> 2;
    if (i < n4) ((float4*)p)[i] = make_float4(0.f, 0.f, 0.f, 0.f);
}

// ---------------- per-edge degree count ----------------
__global__ void edge_count(const int* __restrict__ dst, int E, float* __restrict__ cnt) {
    int e = blockIdx.x * blockDim.x + threadIdx.x;
    if (e < E) unsafeAtomicAdd(cnt + dst[e], 1.0f);
}

// ---------------- edge gather + scatter-add (fp32 atomics, mostly L2-resident) ----------------
__global__ void scatter_feat(const float* __restrict__ xsrc, const int* __restrict__ src,
                             const int* __restrict__ dst, float* __restrict__ agg, int E) {
    int gid = blockIdx.x * blockDim.x + threadIdx.x;
    int e = gid >> 5;
    if (e >= E) return;
    int c = (gid & 31) << 2;                     // 32 lanes x 4 floats = 128 features
    int s = src[e], d = dst[e];
    float4 v = *(const float4*)(xsrc + (long)s * 128 + c);
    float* p = agg + (long)d * 128 + c;
    unsafeAtomicAdd(p + 0, v.x);
    unsafeAtomicAdd(p + 1, v.y);
    unsafeAtomicAdd(p + 2, v.z);
    unsafeAtomicAdd(p + 3, v.w);
}

// ---------------- fused SAGE linear: out = act( (agg/cnt)@Wl^T + x@Wr^T + bias ) ----------------
// One wave computes a 16-row x (NTILE*16)-col stripe with V_WMMA_F32_16X16X4_F32.
// Weights staged in LDS with padded stride. K fixed at 128.
template <bool HAS_AGG, bool RELU, int NTILE>
__global__ __launch_bounds__(256)
void sage_gemm(const float* __restrict__ agg, const float* __restrict__ cnt,
               const float* __restrict__ Wl,
               const float* __restrict__ xb, const float* __restrict__ Wr,
               const float* __restrict__ bias,
               float* __restrict__ out, int n_rows) {
    extern __shared__ float smem[];
    constexpr int NC = NTILE * 16;
    const int tid = threadIdx.x;

    // cooperative LDS staging: Wr at smem[0], Wl at smem[NC*WPAD]
    for (int idx = tid; idx < NC * 128; idx += 256) {
        int r = idx >> 7, c = idx & 127;
        smem[r * WPAD + c] = Wr[idx];
    }
    if constexpr (HAS_AGG) {
        float* s2 = smem + NC * WPAD;
        for (int idx = tid; idx < NC * 128; idx += 256) {
            int r = idx >> 7, c = idx & 127;
            s2[r * WPAD + c] = Wl[idx];
        }
    }
    __syncthreads();

    const int wave = tid >> 5, lane = tid & 31;
    const int l15 = lane & 15, kh = lane >> 4;   // A/B lane mapping: M/N = l15, K-half = kh
    const int rowbase = (blockIdx.x * 8 + wave) * 16;
    if (rowbase >= n_rows) return;

    const float* pb = xb + (long)(rowbase + l15) * 128 + 2 * kh;
    const float* pa = nullptr;
    float s = 0.f;
    if constexpr (HAS_AGG) {
        pa = agg + (long)(rowbase + l15) * 128 + 2 * kh;
        s = 1.0f / fmaxf(cnt[rowbase + l15], 1.0f);   // mean aggregation, guarded
    }

    v8f acc[NTILE] = {};
    const float* smWr = smem;
    const float* smWl = smem + NC * WPAD;

#pragma unroll 4
    for (int kt = 0; kt < 32; ++kt) {
        const int col = kt * 4;
        v2f ab = *(const v2f*)(pb + col);
        v2f al = {};
        if constexpr (HAS_AGG) {
            al = *(const v2f*)(pa + col);
            al = al * s;
        }
        const int bo = col + 2 * kh;
#pragma unroll
        for (int t = 0; t < NTILE; ++t) {
            v2f br = *(const v2f*)(smWr + (t * 16 + l15) * WPAD + bo);
            acc[t] = __builtin_amdgcn_wmma_f32_16x16x4_f32(
                false, ab, false, br, (short)0, acc[t], false, false);
            if constexpr (HAS_AGG) {
                v2f bl2 = *(const v2f*)(smWl + (t * 16 + l15) * WPAD + bo);
                acc[t] = __builtin_amdgcn_wmma_f32_16x16x4_f32(
                    false, al, false, bl2, (short)0, acc[t], false, false);
            }
        }
    }

    // epilogue: bias (+ReLU), D layout: row = r + 8*kh, col = t*16 + l15
#pragma unroll
    for (int t = 0; t < NTILE; ++t) {
        const int j = t * 16 + l15;
        const float bv = bias[j];
#pragma unroll
        for (int r = 0; r < 8; ++r) {
            float v = acc[t][r] + bv;
            if constexpr (RELU) v = fmaxf(v, 0.0f);
            out[(long)(rowbase + r + 8 * kh) * NC + j] = v;
        }
    }
}

// ---------------- driver ----------------
extern "C" void kernel_launch(void* const* d_in, const int* in_sizes, int n_in,
                              void* d_out, int out_size, void* d_ws, size_t ws_size,
                              hipStream_t stream) {
    const float* x_movie = (const float*)d_in[0];
    const float* x_user  = (const float*)d_in[1];
    const int* rev_src   = (const int*)d_in[2];
    const int* rev_dst   = (const int*)d_in[3];
    const int* rates_src = (const int*)d_in[4];
    const int* rates_dst = (const int*)d_in[5];
    const float* Wl1 = (const float*)d_in[6];  const float* bl1 = (const float*)d_in[7];
    const float* Wr1 = (const float*)d_in[8];
    const float* Wl2 = (const float*)d_in[9];  const float* bl2 = (const float*)d_in[10];
    const float* Wr2 = (const float*)d_in[11];
    const float* Wl3 = (const float*)d_in[12]; const float* bl3 = (const float*)d_in[13];
    const float* Wr3 = (const float*)d_in[14];
    const float* W1  = (const float*)d_in[15]; const float* b1  = (const float*)d_in[16];
    const float* W2  = (const float*)d_in[17]; const float* b2  = (const float*)d_in[18];

    const int E       = in_sizes[2];
    const int n_movie = in_sizes[0] / 128;
    const int n_user  = in_sizes[1] / 128;
    float* out = (float*)d_out;

    // workspace layout
    float* ws = (float*)d_ws;
    size_t off = 0;
    float* agg     = ws + off; off += (size_t)n_user * 128;   // reused across convs
    float* user_x  = ws + off; off += (size_t)n_user * 128;
    float* movie_x = ws + off; off += (size_t)n_movie * 128;
    float* cnt_u   = ws + off; off += (size_t)n_user;
    float* cnt_m   = ws + off; off += (size_t)n_movie;

    auto zero = [&](float* p, size_t n) {
        int blocks = (int)((n / 4 + 255) / 256);
        zero_f32<<<blocks, 256, 0, stream>>>(p, (long)n);
    };

    const int eb = (E + 255) / 256;
    const int sb = (int)(((long)E * 32 + 255) / 256);
    const size_t lds2 = (size_t)2 * 128 * WPAD * sizeof(float); // two 128x128 weights
    const size_t lds1 = (size_t)64 * WPAD * sizeof(float);      // one 64x128 weight
    const int gU = (n_user / 16 + 7) / 8;
    const int gM = (n_movie / 16 + 7) / 8;

    // degrees (rev edges reused by conv1 and conv3)
    zero(cnt_u, (size_t)n_user);
    zero(cnt_m, (size_t)n_movie);
    edge_count<<<eb, 256, 0, stream>>>(rev_dst, E, cnt_u);
    edge_count<<<eb, 256, 0, stream>>>(rates_dst, E, cnt_m);

    // conv1: movie -> user
    zero(agg, (size_t)n_user * 128);
    scatter_feat<<<sb, 256, 0, stream>>>(x_movie, rev_src, rev_dst, agg, E);
    sage_gemm<true, true, 8><<<gU, 256, lds2, stream>>>(agg, cnt_u, Wl1, x_user, Wr1, bl1,
                                                        user_x, n_user);

    // conv2: user -> movie
    zero(agg, (size_t)n_movie * 128);
    scatter_feat<<<sb, 256, 0, stream>>>(user_x, rates_src, rates_dst, agg, E);
    sage_gemm<true, true, 8><<<gM, 256, lds2, stream>>>(agg, cnt_m, Wl2, x_movie, Wr2, bl2,
                                                        movie_x, n_movie);

    // conv3: movie -> user (in-place update of user_x; each row owned by one wave)
    zero(agg, (size_t)n_user * 128);
    scatter_feat<<<sb, 256, 0, stream>>>(movie_x, rev_src, rev_dst, agg, E);
    sage_gemm<true, true, 8><<<gU, 256, lds2, stream>>>(agg, cnt_u, Wl3, user_x, Wr3, bl3,
                                                        user_x, n_user);

    // output heads (no relu): d_out = [user_out (n_user x 64) | movie_out (n_movie x 64)]
    sage_gemm<false, false, 4><<<gU, 256, lds1, stream>>>(nullptr, nullptr, nullptr,
                                                          user_x, W1, b1, out, n_user);
    sage_gemm<false, false, 4><<<gM, 256, lds1, stream>>>(nullptr, nullptr, nullptr,
                                                          movie_x, W2, b2,
                                                          out + (size_t)n_user * 64, n_movie);
}